// DiagMean_19232863552217
// MI455X (gfx1250) — compile-verified
//
#include <hip/hip_runtime.h>
#include <hip/hip_bf16.h>

// Problem constants
#define TDIM 2048
#define HALF 1024
#define BINS 2049            // offsets -1024..1024
#define BINS_PAD 2304        // 9 chunks * 256 threads
#define NCHUNK 9
#define NSTRIP 8
#define ROWS_PER_STRIP 256   // 2048 / 8
#define NBATCH 16

typedef __attribute__((ext_vector_type(2))) float v2f;
typedef __attribute__((ext_vector_type(8))) float v8f;

// ---------------------------------------------------------------------------
// Kernel 1: streaming diagonal partial sums.
// Each thread owns one diagonal bin; a wave covers 32 consecutive diagonals so
// every row step is one coalesced 128B global load per wave; accumulation is a
// single register per thread. Writes partial[b][strip][bin] to workspace.
// ---------------------------------------------------------------------------
__global__ __launch_bounds__(256) void diagmean_partial(
    const float* __restrict__ in, float* __restrict__ part) {
  const int chunk = blockIdx.x;   // 0..8   (256 bins each)
  const int strip = blockIdx.y;   // 0..7   (256 rows each)
  const int b     = blockIdx.z;   // 0..15
  const int bin   = chunk * 256 + (int)threadIdx.x;   // 0..2303
  const int d     = bin - HALF;                       // diagonal offset

  float acc = 0.0f;
  if (d <= HALF) {  // bins > 2048 are padding: keep acc = 0
    const float* pb = in + (size_t)b * TDIM * TDIM;
    int i0 = strip * ROWS_PER_STRIP;
    // valid i: i >= max(0,-d), i <= T-2, j = i+d <= T-2
    int ilo = i0;
    if (d < 0 && ilo < -d) ilo = -d;
    int ihi = i0 + ROWS_PER_STRIP;
    if (ihi > TDIM - 1) ihi = TDIM - 1;          // i <= 2046
    int jcap = (TDIM - 1) - d;                   // i <= 2046 - d (exclusive 2047-d)
    if (d > 0 && ihi > jcap) ihi = jcap;

    size_t idx = (size_t)((long)ilo * (TDIM + 1) + d);  // i*T + (i+d)
#pragma unroll 4
    for (int i = ilo; i < ihi; ++i) {
      acc += pb[idx];
      idx += (TDIM + 1);
    }
  }
  part[((size_t)b * NSTRIP + strip) * BINS_PAD + bin] = acc;
}

// ---------------------------------------------------------------------------
// Kernel 2: reduce 8 strips per bin with V_WMMA_F32_16X16X4_F32 (A = ones, so
// D[m][n] = sum_k B[k][n] for every lane), then mean / center / negate.
// One block per batch, 4 waves; each wave reduces 16-bin tiles.
// ---------------------------------------------------------------------------
__global__ __launch_bounds__(128) void diagmean_finalize(
    const float* __restrict__ part, float* __restrict__ out) {
  const int b    = blockIdx.x;
  const int tid  = (int)threadIdx.x;
  const int lane = tid & 31;
  const int wave = tid >> 5;          // 0..3
  const int col  = lane & 15;         // B-matrix column (bin within tile)
  const int hi   = lane >> 4;         // which K-row pair this lane holds

  __shared__ float means[2064];
  __shared__ float red[128];
  __shared__ float gmean_s;

  const float* pb = part + (size_t)b * NSTRIP * BINS_PAD;

  v2f a; a.x = 1.0f; a.y = 1.0f;      // all-ones A (16x4)

  const int NTILES = 129;             // ceil(2049 / 16)
  for (int tile = wave; tile < NTILES; tile += 4) {
    const int bin = tile * 16 + col;  // <= 2063 < BINS_PAD, always in-bounds
    // First WMMA reduces strips {0,1,2,3}; second reduces {4,5,6,7}.
    v2f b0, b1;
    b0.x = pb[(size_t)(2 * hi + 0) * BINS_PAD + bin];
    b0.y = pb[(size_t)(2 * hi + 1) * BINS_PAD + bin];
    b1.x = pb[(size_t)(4 + 2 * hi) * BINS_PAD + bin];
    b1.y = pb[(size_t)(5 + 2 * hi) * BINS_PAD + bin];

    v8f c = {};
    c = __builtin_amdgcn_wmma_f32_16x16x4_f32(false, a, false, b0,
                                              (short)0, c, false, false);
    c = __builtin_amdgcn_wmma_f32_16x16x4_f32(false, a, false, b1,
                                              (short)0, c, false, false);
    // Every lane's c[0] is the column sum for its bin; lanes 0-15 publish.
    if (hi == 0 && bin < BINS) {
      int dd = bin - HALF;
      float cnt = (float)((TDIM - 1) - (dd < 0 ? -dd : dd));
      means[bin] = c[0] / cnt;
    }
  }
  __syncthreads();

  // Deterministic reduction of the 2049 means for centering.
  float p = 0.0f;
  for (int i = tid; i < BINS; i += 128) p += means[i];
  red[tid] = p;
  __syncthreads();
  if (tid == 0) {
    float s = 0.0f;
    for (int i = 0; i < 128; ++i) s += red[i];
    gmean_s = s / (float)BINS;
  }
  __syncthreads();

  const float g = gmean_s;
  for (int i = tid; i < BINS; i += 128) {
    out[(size_t)b * BINS + i] = -(means[i] - g);
  }
}

extern "C" void kernel_launch(void* const* d_in, const int* in_sizes, int n_in,
                              void* d_out, int out_size, void* d_ws, size_t ws_size,
                              hipStream_t stream) {
  (void)in_sizes; (void)n_in; (void)out_size; (void)ws_size;
  const float* in  = (const float*)d_in[0];
  float*       out = (float*)d_out;
  float*       ws  = (float*)d_ws;   // needs 16*8*2304*4 = ~1.13 MB

  dim3 g1(NCHUNK, NSTRIP, NBATCH);
  diagmean_partial<<<g1, 256, 0, stream>>>(in, ws);
  diagmean_finalize<<<NBATCH, 128, 0, stream>>>(ws, out);
}